// BinEmbedding_49520972923592
// MI455X (gfx1250) — compile-verified
//
#include <hip/hip_runtime.h>

// BinEmbedding forward for MI455X (gfx1250, wave32).
// out[row, :] = emb_table[tok(row)], tok = isnan(x) ? 0 : max(searchsorted_right(bins,x), 1)
//
// Memory-bound: 256 MB streamed out, ~4 MB in -> ~11 us floor at 23.3 TB/s.
// Strategy: TDM (tensor_load_to_lds) parks the 64.3 KB table in LDS once per
// block; bins (1 KB) also in LDS; 16 lanes/row emit one non-temporal 128-bit
// store each (512 contiguous bytes per wave32).

typedef float  f4  __attribute__((ext_vector_type(4)));
typedef unsigned int v4u __attribute__((ext_vector_type(4)));
typedef int    v4i __attribute__((ext_vector_type(4)));
typedef int    v8i __attribute__((ext_vector_type(8)));

#define NBINS   256
#define HDIM    64
#define TROWS   257                    // 256 bins + mask token row 0
#define ROWS    (16 * 65536)           // B * L
#define CHUNKS  (ROWS * (HDIM / 4))    // float4 chunks of output
#define BLK     256
#define GRID    1024

__global__ __launch_bounds__(BLK) void binemb_gather_kernel(
    const float* __restrict__ x,
    const float* __restrict__ bins,
    const float* __restrict__ table,
    f4* __restrict__ out)
{
    __shared__ f4    s_table[TROWS * (HDIM / 4)];   // 65,792 B
    __shared__ float s_bins[NBINS];                 //  1,024 B

    // bins: one float per thread (blockDim == 256)
    s_bins[threadIdx.x] = bins[threadIdx.x];

    // Table: Tensor Data Mover, issued once by wave 0 (TDM ignores EXEC,
    // issues per-wave). 1-D descriptor: 16448 elements of 4 bytes.
    if (threadIdx.x < 32) {
        unsigned lds_off = (unsigned)(size_t)(void*)&s_table[0]; // LDS byte offset = addr[31:0]
        unsigned long long ga = (unsigned long long)(size_t)table;
        const unsigned NE = TROWS * HDIM;           // 16448 elements

        // D# group 0: count=1 | lds_addr | global_addr[31:0] | global_addr[56:32], type=2
        v4u g0 = { 1u,
                   lds_off,
                   (unsigned)(ga & 0xFFFFFFFFu),
                   (unsigned)((ga >> 32) & 0x01FFFFFFu) | 0x80000000u };
        // D# group 1: data_size=2 (4B); tensor_dim0=NE; tensor_dim1=1;
        //             tile_dim0=NE; tile_dim1=1; tensor_dim0_stride=NE
        v8i g1 = { (int)0x00020000u,                 // wg_mask=0, data_size=4B
                   (int)(NE << 16),                  // tensor_dim0[15:0] in [31:16]
                   (int)((NE >> 16) | (1u << 16)),   // tensor_dim0[31:16] | tensor_dim1[15:0]=1
                   (int)(NE << 16),                  // tensor_dim1 hi=0 | tile_dim0=NE
                   (int)1,                           // tile_dim1=1, tile_dim2=0
                   (int)NE,                          // tensor_dim0_stride[31:0]
                   0, 0 };                           // stride hi / dim1_stride = 0
        v4i gz4 = { 0, 0, 0, 0 };                    // groups 2/3 unused (<=2D)
        v8i gz8 = { 0, 0, 0, 0, 0, 0, 0, 0 };        // trailing group (clang-23 6-arg form)

        __builtin_amdgcn_tensor_load_to_lds(g0, g1, gz4, gz4, gz8, 0);
        __builtin_amdgcn_s_wait_tensorcnt(0);
    }
    __syncthreads();

    const int stride = GRID * BLK;
    int c = blockIdx.x * BLK + threadIdx.x;
    for (; c < CHUNKS; c += stride) {
        const int row = c >> 4;        // output row (b*L + l)
        const int sub = c & 15;        // which float4 of the 64-float row

        const float v = x[row];        // 16 lanes broadcast-load the same word
        int tok;
        if (v != v) {                  // NaN -> mask token 0
            tok = 0;
        } else {
            // searchsorted_right over 256 sorted bins: 9-step binary search in LDS
            int cnt = 0;
            #pragma unroll
            for (int s = 256; s >= 1; s >>= 1) {
                const int np = cnt + s;
                if (np <= NBINS && s_bins[np - 1] <= v) cnt = np;
            }
            tok = (cnt < 1) ? 1 : cnt; // max(searchsorted-1, 0) + 1
        }

        const f4 val = s_table[tok * (HDIM / 4) + sub]; // ds_load_b128, conflict-free
        __builtin_nontemporal_store(val, &out[c]);      // streaming 256 MB: TH=NT
    }
}

extern "C" void kernel_launch(void* const* d_in, const int* in_sizes, int n_in,
                              void* d_out, int out_size, void* d_ws, size_t ws_size,
                              hipStream_t stream) {
    const float* x     = (const float*)d_in[0];  // (16, 65536) f32, ~10% NaN
    const float* bins  = (const float*)d_in[1];  // (256,) f32 sorted
    const float* table = (const float*)d_in[2];  // (257, 64) f32
    f4* out = (f4*)d_out;                        // (16, 65536, 64) f32

    (void)in_sizes; (void)n_in; (void)out_size; (void)d_ws; (void)ws_size;

    binemb_gather_kernel<<<dim3(GRID), dim3(BLK), 0, stream>>>(x, bins, table, out);
}